// ChannelMultiplier_58600533786808
// MI455X (gfx1250) — compile-verified
//
#include <hip/hip_runtime.h>

// out[n,c,h,w] = x[n,c,h,w] * multiplier[c]
// N=32, C=256, H=W=56  ->  HW = 3136 floats = 784 float4 per (n,c) plane.
// Pure HBM-bound stream (AI = 0.125 FLOP/B): optimize bytes, not FLOPs.

typedef __attribute__((ext_vector_type(4))) float fvec4;

#ifndef CM_N
#define CM_N 32
#define CM_C 256
#define CM_HW4 784   // (56*56)/4 float4 elements per channel plane
#endif

__launch_bounds__(256)
__global__ void ChannelMultiplier_58600533786808_kernel(
    const float* __restrict__ x,
    const float* __restrict__ multiplier,
    float* __restrict__ out)
{
    // One block per (n, c) plane: blockIdx.x = n*C + c, layout is NCHW contiguous.
    const unsigned plane = blockIdx.x;
    const unsigned c     = plane & (CM_C - 1);   // wave-uniform -> s_load_b32
    const float    s     = multiplier[c];

    const fvec4* __restrict__ xin = reinterpret_cast<const fvec4*>(x)
                                    + (size_t)plane * CM_HW4;
    fvec4* __restrict__ o         = reinterpret_cast<fvec4*>(out)
                                    + (size_t)plane * CM_HW4;

    const unsigned t = threadIdx.x;              // blockDim.x == 256 (8 wave32s)

    // 784 = 3*256 + 16: three full strips + a 16-lane tail.
    // Issue all non-temporal 128-bit loads first for max memory-level parallelism.
    fvec4 v0 = __builtin_nontemporal_load(xin + t);
    fvec4 v1 = __builtin_nontemporal_load(xin + t + 256);
    fvec4 v2 = __builtin_nontemporal_load(xin + t + 512);
    const bool tail = (t < (CM_HW4 - 768));      // t < 16
    fvec4 v3;
    if (tail) v3 = __builtin_nontemporal_load(xin + t + 768);

    v0 *= s;
    v1 *= s;
    v2 *= s;

    __builtin_nontemporal_store(v0, o + t);
    __builtin_nontemporal_store(v1, o + t + 256);
    __builtin_nontemporal_store(v2, o + t + 512);
    if (tail) {
        v3 *= s;
        __builtin_nontemporal_store(v3, o + t + 768);
    }
}

extern "C" void kernel_launch(void* const* d_in, const int* in_sizes, int n_in,
                              void* d_out, int out_size, void* d_ws, size_t ws_size,
                              hipStream_t stream)
{
    (void)in_sizes; (void)n_in; (void)out_size; (void)d_ws; (void)ws_size;

    const float* x    = reinterpret_cast<const float*>(d_in[0]);
    const float* mult = reinterpret_cast<const float*>(d_in[1]);
    float*       out  = reinterpret_cast<float*>(d_out);

    // 32*256 = 8192 blocks of 256 threads (8 wave32 each).
    dim3 grid(CM_N * CM_C);
    dim3 block(256);
    ChannelMultiplier_58600533786808_kernel<<<grid, block, 0, stream>>>(x, mult, out);
}